// VectorQuantizer_87144886435978
// MI455X (gfx1250) — compile-verified
//
#include <hip/hip_runtime.h>
#include <math.h>

typedef float v2f __attribute__((ext_vector_type(2)));
typedef float v8f __attribute__((ext_vector_type(8)));

#define K_CODES   1024
#define D_DIM     256
#define N_ROWS    32768
#define ROWS_WG   16
#define ZT_STRIDE 258          // 256 + 2 pad (banks: 258 % 64 == 2)
#define DS_STRIDE 1026         // 1024 + 2 pad
#define INV_TEMP  20.0f        // 1 / 0.05

// ---------------------------------------------------------------------------
// Kernel 0: zero the atomic accumulators in d_out (loss scalar + avg probs)
// ---------------------------------------------------------------------------
__global__ void vq_init(float* __restrict__ out) {
  int i = blockIdx.x * blockDim.x + threadIdx.x;
  float* loss = out + (size_t)N_ROWS * D_DIM;          // offset 8388608
  if (i == 0) loss[0] = 0.0f;
  float* avg = loss + 1 + N_ROWS;                      // offset 8421377
  if (i < K_CODES) avg[i] = 0.0f;
}

// ---------------------------------------------------------------------------
// Kernel 1: esq[k] = ||emb_k||^2   (emb is 1024 x 256, L2-resident)
// ---------------------------------------------------------------------------
__global__ void vq_esq(const float* __restrict__ emb, float* __restrict__ esq) {
  int k = blockIdx.x * blockDim.x + threadIdx.x;
  if (k < K_CODES) {
    const float* e = emb + (size_t)k * D_DIM;
    float s = 0.0f;
    for (int d = 0; d < D_DIM; ++d) s += e[d] * e[d];
    esq[k] = s;
  }
}

// ---------------------------------------------------------------------------
// Kernel 2: fused distances (WMMA f32) + softmax-mean + argmin + gather + loss
// One workgroup = 16 rows of z. 256 threads = 8 waves (wave32).
// Loop order: kc outer, tile inner -> A fragment loaded once per kc and
// reused by 8 WMMAs into 8 resident v8f accumulators.
// ---------------------------------------------------------------------------
struct SmemT {
  float zt[ROWS_WG * ZT_STRIDE];   // at offset 0: DS immediate offsets fit
  float dist[ROWS_WG * DS_STRIDE];
  float esq_s[K_CODES];
  float zsq[ROWS_WG];
};

__global__ __launch_bounds__(256) void vq_main(const float* __restrict__ z,
                                               const float* __restrict__ emb,
                                               const float* __restrict__ esq,
                                               float* __restrict__ out) {
  __shared__ SmemT sm;

  const int tid  = threadIdx.x;
  const int wave = tid >> 5;
  const int lane = tid & 31;
  const int hv   = lane >> 4;    // half-wave (selects K/M sub-block in frags)
  const int m    = lane & 15;
  const int row0 = blockIdx.x * ROWS_WG;

  // --- stage z tile + ||e||^2 into LDS ------------------------------------
  for (int i = tid; i < ROWS_WG * D_DIM; i += 256) {
    int r = i >> 8, c = i & 255;
    sm.zt[r * ZT_STRIDE + c] = z[(size_t)(row0 + r) * D_DIM + c];
  }
  for (int i = tid; i < K_CODES; i += 256) sm.esq_s[i] = esq[i];
  __syncthreads();

  if (tid < ROWS_WG) {
    float s = 0.0f;
    for (int c = 0; c < D_DIM; ++c) { float v = sm.zt[tid * ZT_STRIDE + c]; s += v * v; }
    sm.zsq[tid] = s;
  }
  __syncthreads();

  // --- WMMA phase: wave w owns column tiles [w*128, w*128+128) -------------
  // A (16x4 f32): lane (m,hv) holds z[m][4kc+2hv .. +1]
  // B (4x16 f32): lane (m,hv) holds emb[colt+m][4kc+2hv .. +1]  (B = emb^T)
  // C (16x16 f32): VGPR r, lane (m,hv) = G[r + 8*hv][m]
  // One B base pointer: tile t is a constant +16*D_DIM stride (imm offset).
  const float* eb = emb + (size_t)(wave * 128 + m) * D_DIM + 2 * hv;
  const float* zb = &sm.zt[m * ZT_STRIDE + 2 * hv];

  v8f acc[8];
#pragma unroll
  for (int t = 0; t < 8; ++t) acc[t] = (v8f){};

  for (int kc = 0; kc < 64; ++kc) {
    v2f a = *(const v2f*)(zb + 4 * kc);
#pragma unroll
    for (int t = 0; t < 8; ++t) {
      v2f b = *(const v2f*)(eb + t * (16 * D_DIM) + 4 * kc);
      acc[t] = __builtin_amdgcn_wmma_f32_16x16x4_f32(
          /*neg_a=*/false, a, /*neg_b=*/false, b,
          /*c_mod=*/(short)0, acc[t], /*reuse_a=*/false, /*reuse_b=*/false);
    }
  }

#pragma unroll
  for (int t = 0; t < 8; ++t) {
    const int col0 = (wave * 8 + t) * 16;
    const float ecol = sm.esq_s[col0 + m];
#pragma unroll
    for (int r = 0; r < 8; ++r) {
      const int rr = r + 8 * hv;
      sm.dist[rr * DS_STRIDE + col0 + m] = sm.zsq[rr] + ecol - 2.0f * acc[t][r];
    }
  }
  __syncthreads();

  // --- per-row epilogue: argmin, softmax mean, gather, loss ----------------
  const float invN      = 1.0f / (float)N_ROWS;
  const float lossScale = 1.25f / (float)((size_t)N_ROWS * D_DIM);
  float* q_out    = out;
  float* loss_out = out + (size_t)N_ROWS * D_DIM;
  float* idx_out  = loss_out + 1;
  float* avg_out  = idx_out + N_ROWS;

  for (int rl = wave * 2; rl < wave * 2 + 2; ++rl) {
    const float* drow = &sm.dist[rl * DS_STRIDE];

    // argmin with first-index tie-break (matches jnp.argmin)
    float minv = 3.402823e38f; int mini = 0;
    for (int k = lane; k < K_CODES; k += 32) {
      float d = drow[k];
      if (d < minv) { minv = d; mini = k; }
    }
    for (int off = 16; off > 0; off >>= 1) {
      float ov = __shfl_xor(minv, off, 32);
      int   oi = __shfl_xor(mini, off, 32);
      if (ov < minv || (ov == minv && oi < mini)) { minv = ov; mini = oi; }
    }

    // softmax(-d/T): denominator with max-logit shift (= exp((dmin-d)/T))
    float ssum = 0.0f;
    for (int k = lane; k < K_CODES; k += 32)
      ssum += __expf((minv - drow[k]) * INV_TEMP);
    for (int off = 16; off > 0; off >>= 1) ssum += __shfl_xor(ssum, off, 32);
    const float pscale = invN / ssum;
    for (int k = lane; k < K_CODES; k += 32)
      atomicAdd(&avg_out[k], __expf((minv - drow[k]) * INV_TEMP) * pscale);

    // straight-through output = emb[argmin]; vq_loss = 1.25 * MSE(q, z)
    const int grow = row0 + rl;
    const float* erow = emb + (size_t)mini * D_DIM;
    float sq = 0.0f;
    for (int d = lane; d < D_DIM; d += 32) {
      float q  = erow[d];
      float zv = sm.zt[rl * ZT_STRIDE + d];
      float df = q - zv;
      sq += df * df;
      q_out[(size_t)grow * D_DIM + d] = q;
    }
    for (int off = 16; off > 0; off >>= 1) sq += __shfl_xor(sq, off, 32);
    if (lane == 0) {
      atomicAdd(loss_out, sq * lossScale);
      idx_out[grow] = (float)mini;
    }
  }
}

// ---------------------------------------------------------------------------
extern "C" void kernel_launch(void* const* d_in, const int* in_sizes, int n_in,
                              void* d_out, int out_size, void* d_ws, size_t ws_size,
                              hipStream_t stream) {
  const float* z   = (const float*)d_in[0];   // (32,1024,256) f32
  const float* emb = (const float*)d_in[1];   // (1024,256)    f32
  float* out = (float*)d_out;
  float* esq = (float*)d_ws;                  // 1024 floats of scratch

  vq_init<<<4, 256, 0, stream>>>(out);
  vq_esq <<<4, 256, 0, stream>>>(emb, esq);
  vq_main<<<N_ROWS / ROWS_WG, 256, 0, stream>>>(z, emb, esq, out);
}